// TransformParamsLayer_90168543412828
// MI455X (gfx1250) — compile-verified
//
#include <hip/hip_runtime.h>

// 2-D similarity-transform (Procrustes) fit, B independent rows.
// Memory-bound streaming reduction: 128 MiB in -> 384 KiB out (~5.5us at HBM
// peak; L2-resident across timed graph replays since 128 MiB < 192 MB L2).
// One wave32 per row; coalesced global_load_b128 streams; wave32 xor-shuffle
// tree reduction (ds ops); global_prefetch_b8 to overlap the second stream.

template <int NCHUNK>  // float4 chunks per row (N pairs = 2*NCHUNK)
__global__ __launch_bounds__(256) void tp_kernel_fixed(
    const float* __restrict__ x,   // (B, 2N)
    const float* __restrict__ ms,  // (B, N, 2)
    float* __restrict__ out,       // (B, 6)
    int B)
{
    const int gtid = blockIdx.x * blockDim.x + threadIdx.x;
    const int row  = gtid >> 5;   // one wave32 per row
    const int lane = gtid & 31;
    if (row >= B) return;

    const float4* __restrict__ xr = reinterpret_cast<const float4*>(x)  + (size_t)row * NCHUNK;
    const float4* __restrict__ mr = reinterpret_cast<const float4*>(ms) + (size_t)row * NCHUNK;

    // Warm GL2/WGP$ for the ms stream (its loads trail the x loads in program
    // order); lowers to global_prefetch_b8, one cacheline probe per lane.
    __builtin_prefetch(&mr[lane], 0, 0);
    __builtin_prefetch(&mr[lane + NCHUNK / 2], 0, 0);

    float Sx = 0.f, Sy = 0.f, Dx = 0.f, Dy = 0.f;
    float SS = 0.f, SD = 0.f, CR = 0.f;

#pragma unroll
    for (int i = 0; i < NCHUNK / 32; ++i) {
        const float4 s = xr[lane + i * 32];   // (sx0, sy0, sx1, sy1)
        const float4 d = mr[lane + i * 32];   // (dx0, dy0, dx1, dy1)
        Sx += s.x + s.z;   Sy += s.y + s.w;
        Dx += d.x + d.z;   Dy += d.y + d.w;
        SS = fmaf(s.x, s.x, fmaf(s.y, s.y, fmaf(s.z, s.z, fmaf(s.w, s.w, SS))));
        SD = fmaf(s.x, d.x, fmaf(s.y, d.y, fmaf(s.z, d.z, fmaf(s.w, d.w, SD))));
        CR = fmaf(s.x, d.y, fmaf(-s.y, d.x, fmaf(s.z, d.w, fmaf(-s.w, d.z, CR))));
    }

    // wave32 tree reduction (5 xor-shuffle steps, ds_swizzle path)
#pragma unroll
    for (int m = 16; m >= 1; m >>= 1) {
        Sx += __shfl_xor(Sx, m, 32);
        Sy += __shfl_xor(Sy, m, 32);
        Dx += __shfl_xor(Dx, m, 32);
        Dy += __shfl_xor(Dy, m, 32);
        SS += __shfl_xor(SS, m, 32);
        SD += __shfl_xor(SD, m, 32);
        CR += __shfl_xor(CR, m, 32);
    }

    if (lane == 0) {
        const float invN = 0.5f / (float)NCHUNK;          // 1/N, N = 2*NCHUNK pairs
        const float norm = SS - (Sx * Sx + Sy * Sy) * invN;
        const float rinv = 1.0f / norm;
        const float a = (SD - (Sx * Dx + Sy * Dy) * invN) * rinv;
        const float b = (CR - (Sx * Dy - Sy * Dx) * invN) * rinv;
        const float mx  = Sx * invN, my  = Sy * invN;
        const float mdx = Dx * invN, mdy = Dy * invN;
        const float tx = mdx - (mx * a - my * b);
        const float ty = mdy - (mx * b + my * a);
        float2* o = reinterpret_cast<float2*>(out + (size_t)row * 6);  // 24B row stride, 8B aligned
        o[0] = make_float2(a, b);
        o[1] = make_float2(-b, a);
        o[2] = make_float2(tx, ty);
    }
}

// generic fallback for other row lengths (same algorithm, runtime chunk count)
__global__ __launch_bounds__(256) void tp_kernel_dyn(
    const float* __restrict__ x, const float* __restrict__ ms,
    float* __restrict__ out, int B, int nchunk)
{
    const int gtid = blockIdx.x * blockDim.x + threadIdx.x;
    const int row  = gtid >> 5;
    const int lane = gtid & 31;
    if (row >= B) return;

    const float4* __restrict__ xr = reinterpret_cast<const float4*>(x)  + (size_t)row * nchunk;
    const float4* __restrict__ mr = reinterpret_cast<const float4*>(ms) + (size_t)row * nchunk;

    float Sx = 0.f, Sy = 0.f, Dx = 0.f, Dy = 0.f;
    float SS = 0.f, SD = 0.f, CR = 0.f;

    for (int i = lane; i < nchunk; i += 32) {
        const float4 s = xr[i];
        const float4 d = mr[i];
        Sx += s.x + s.z;   Sy += s.y + s.w;
        Dx += d.x + d.z;   Dy += d.y + d.w;
        SS = fmaf(s.x, s.x, fmaf(s.y, s.y, fmaf(s.z, s.z, fmaf(s.w, s.w, SS))));
        SD = fmaf(s.x, d.x, fmaf(s.y, d.y, fmaf(s.z, d.z, fmaf(s.w, d.w, SD))));
        CR = fmaf(s.x, d.y, fmaf(-s.y, d.x, fmaf(s.z, d.w, fmaf(-s.w, d.z, CR))));
    }

#pragma unroll
    for (int m = 16; m >= 1; m >>= 1) {
        Sx += __shfl_xor(Sx, m, 32);
        Sy += __shfl_xor(Sy, m, 32);
        Dx += __shfl_xor(Dx, m, 32);
        Dy += __shfl_xor(Dy, m, 32);
        SS += __shfl_xor(SS, m, 32);
        SD += __shfl_xor(SD, m, 32);
        CR += __shfl_xor(CR, m, 32);
    }

    if (lane == 0) {
        const float invN = 0.5f / (float)nchunk;
        const float norm = SS - (Sx * Sx + Sy * Sy) * invN;
        const float rinv = 1.0f / norm;
        const float a = (SD - (Sx * Dx + Sy * Dy) * invN) * rinv;
        const float b = (CR - (Sx * Dy - Sy * Dx) * invN) * rinv;
        const float mx  = Sx * invN, my  = Sy * invN;
        const float mdx = Dx * invN, mdy = Dy * invN;
        const float tx = mdx - (mx * a - my * b);
        const float ty = mdy - (mx * b + my * a);
        float2* o = reinterpret_cast<float2*>(out + (size_t)row * 6);
        o[0] = make_float2(a, b);
        o[1] = make_float2(-b, a);
        o[2] = make_float2(tx, ty);
    }
}

extern "C" void kernel_launch(void* const* d_in, const int* in_sizes, int n_in,
                              void* d_out, int out_size, void* d_ws, size_t ws_size,
                              hipStream_t stream)
{
    const float* x  = (const float*)d_in[0];   // (B, 2N) float32
    const float* ms = (const float*)d_in[1];   // (B, N, 2) float32
    float* out = (float*)d_out;                // (B, 6) float32

    const int B = out_size / 6;                        // 16384
    const int rowFloats = in_sizes[0] / B;             // 2N = 1024
    const int nchunk = rowFloats / 4;                  // float4 per row = 256

    const int rowsPerBlock = 256 / 32;                 // 8 waves -> 8 rows per block
    const int grid = (B + rowsPerBlock - 1) / rowsPerBlock;

    if (nchunk == 256) {
        tp_kernel_fixed<256><<<grid, 256, 0, stream>>>(x, ms, out, B);
    } else {
        tp_kernel_dyn<<<grid, 256, 0, stream>>>(x, ms, out, B, nchunk);
    }
}